// GSAPool_25847113187909
// MI455X (gfx1250) — compile-verified
//
#include <hip/hip_runtime.h>

typedef __attribute__((ext_vector_type(2))) float v2f;
typedef __attribute__((ext_vector_type(4))) float v4f;
typedef __attribute__((ext_vector_type(8))) float v8f;

#define BB 64
#define MM 1024
#define NN 1024
#define DD 256
#define EPSF 1e-12f

#define TM 128
#define TN 64
#define KC 16
#define XSTR 20   // padded LDS row stride (floats): 80B rows -> 16B aligned, conflict-free b64 reads
#define YSTR 20

// ---------------- Kernel 1: squared row norms (one wave per row) ----------------
__global__ __launch_bounds__(256) void row_norms(const float* __restrict__ x,
                                                 const float* __restrict__ y,
                                                 float* __restrict__ xx,
                                                 float* __restrict__ yy) {
  int wid  = (int)((blockIdx.x * blockDim.x + threadIdx.x) >> 5);
  int lane = threadIdx.x & 31;
  const float* src;
  float* dst;
  int row;
  if (wid < BB * MM) { src = x; dst = xx; row = wid; }
  else               { src = y; dst = yy; row = wid - BB * MM; }
  const v4f* p = (const v4f*)(src + (size_t)row * DD);
  v4f v0 = p[lane];        // 256 floats / 32 lanes = 8 per lane
  v4f v1 = p[lane + 32];
  float s = v0.x * v0.x + v0.y * v0.y + v0.z * v0.z + v0.w * v0.w
          + v1.x * v1.x + v1.y * v1.y + v1.z * v1.z + v1.w * v1.w;
  for (int o = 16; o > 0; o >>= 1) s += __shfl_down(s, o, 32);
  if (lane == 0) dst[row] = s;
}

// ---------------- Kernel 2: zero per-batch sum accumulators ----------------
__global__ void zero_sums(float* __restrict__ sums) {
  sums[threadIdx.x] = 0.0f;
}

// ---------------- Kernel 3: WMMA f32 distance GEMM + dist write + batch sum ----------------
__global__ __launch_bounds__(256) void dist_gemm(const float* __restrict__ x,
                                                 const float* __restrict__ y,
                                                 const float* __restrict__ xx,
                                                 const float* __restrict__ yy,
                                                 float* __restrict__ dist,
                                                 float* __restrict__ sums) {
  __shared__ float lx[TM * XSTR];   // 10240 B
  __shared__ float ly[TN * YSTR];   //  5120 B
  __shared__ float wsum[8];

  const int b   = blockIdx.z;
  const int m0  = blockIdx.y * TM;
  const int n0  = blockIdx.x * TN;
  const int tid = threadIdx.x;
  const int lane = tid & 31;
  const int wid  = tid >> 5;   // 0..7
  const int wm   = wid >> 1;   // 0..3  (M direction, 32 rows each)
  const int wn   = wid & 1;    // 0..1  (N direction, 32 cols each)

  const float* xb = x + (size_t)b * MM * DD;
  const float* yb = y + (size_t)b * NN * DD;

  v8f c00 = {}, c01 = {}, c10 = {}, c11 = {};

  const int frow = lane & 15;          // fragment row within 16-tile
  const int fk   = (lane >> 4) * 2;    // lanes 16-31 hold K+2,K+3

  for (int k0 = 0; k0 < DD; k0 += KC) {
    // cooperative global -> LDS: X chunk 128x16, Y chunk 64x16 (float4 per thread)
    {
      int r   = tid >> 2;      // 0..63
      int seg = tid & 3;       // 4 float4 per 16-float row
      v4f vx0 = *(const v4f*)(xb + (size_t)(m0 + r) * DD + k0 + seg * 4);
      *(v4f*)(&lx[r * XSTR + seg * 4]) = vx0;
      v4f vx1 = *(const v4f*)(xb + (size_t)(m0 + r + 64) * DD + k0 + seg * 4);
      *(v4f*)(&lx[(r + 64) * XSTR + seg * 4]) = vx1;
      v4f vy0 = *(const v4f*)(yb + (size_t)(n0 + r) * DD + k0 + seg * 4);
      *(v4f*)(&ly[r * YSTR + seg * 4]) = vy0;
    }
    __syncthreads();
#pragma unroll
    for (int kk = 0; kk < KC; kk += 4) {
      v2f a0 = *(const v2f*)(&lx[(wm * 32 +      frow) * XSTR + kk + fk]);
      v2f a1 = *(const v2f*)(&lx[(wm * 32 + 16 + frow) * XSTR + kk + fk]);
      v2f b0 = *(const v2f*)(&ly[(wn * 32 +      frow) * YSTR + kk + fk]);
      v2f b1 = *(const v2f*)(&ly[(wn * 32 + 16 + frow) * YSTR + kk + fk]);
      c00 = __builtin_amdgcn_wmma_f32_16x16x4_f32(false, a0, false, b0, (short)0, c00, false, false);
      c01 = __builtin_amdgcn_wmma_f32_16x16x4_f32(false, a0, false, b1, (short)0, c01, false, false);
      c10 = __builtin_amdgcn_wmma_f32_16x16x4_f32(false, a1, false, b0, (short)0, c10, false, false);
      c11 = __builtin_amdgcn_wmma_f32_16x16x4_f32(false, a1, false, b1, (short)0, c11, false, false);
    }
    __syncthreads();
  }

  // Epilogue: C/D layout -> VGPR e, lanes 0-15: M=e, N=lane; lanes 16-31: M=8+e, N=lane-16
  const float* xxb = xx + b * MM;
  const float* yyb = yy + b * NN;
  float* db = dist + (size_t)b * MM * NN;
  const int rowh = (lane >> 4) * 8;
  const int coln = lane & 15;
  float lsum = 0.0f;

#define EPILOGUE(CC, I, J)                                            \
  {                                                                   \
    int mbase = m0 + wm * 32 + (I) * 16 + rowh;                       \
    int nn    = n0 + wn * 32 + (J) * 16 + coln;                       \
    float yv  = yyb[nn];                                              \
    _Pragma("unroll")                                                 \
    for (int e = 0; e < 8; ++e) {                                     \
      int mm  = mbase + e;                                            \
      float d2 = xxb[mm] + yv - 2.0f * (CC)[e];                       \
      float dd = sqrtf(fmaxf(d2, EPSF));                              \
      db[(size_t)mm * NN + nn] = dd;                                  \
      lsum += dd;                                                     \
    }                                                                 \
  }
  EPILOGUE(c00, 0, 0)
  EPILOGUE(c01, 0, 1)
  EPILOGUE(c10, 1, 0)
  EPILOGUE(c11, 1, 1)
#undef EPILOGUE

  for (int o = 16; o > 0; o >>= 1) lsum += __shfl_down(lsum, o, 32);
  if (lane == 0) wsum[wid] = lsum;
  __syncthreads();
  if (tid == 0) {
    float s = 0.0f;
#pragma unroll
    for (int w = 0; w < 8; ++w) s += wsum[w];
    atomicAdd(&sums[b], s);
  }
}

// ---------------- Kernel 4: mean-threshold mask (float4 streaming) ----------------
__global__ __launch_bounds__(256) void mask_kernel(const float* __restrict__ dist,
                                                   const float* __restrict__ sums,
                                                   float* __restrict__ mask) {
  size_t i4 = (size_t)blockIdx.x * blockDim.x + threadIdx.x;
  int b = (int)(i4 >> 18);  // 1M elems per batch -> 256K float4 per batch
  float avg = sums[b] * (1.0f / (1024.0f * 1024.0f));
  v4f d = ((const v4f*)dist)[i4];
  v4f m;
  m.x = (d.x <= avg) ? 1.0f : 0.0f;
  m.y = (d.y <= avg) ? 1.0f : 0.0f;
  m.z = (d.z <= avg) ? 1.0f : 0.0f;
  m.w = (d.w <= avg) ? 1.0f : 0.0f;
  ((v4f*)mask)[i4] = m;
}

extern "C" void kernel_launch(void* const* d_in, const int* in_sizes, int n_in,
                              void* d_out, int out_size, void* d_ws, size_t ws_size,
                              hipStream_t stream) {
  const float* x = (const float*)d_in[0];
  const float* y = (const float*)d_in[1];
  float* dist = (float*)d_out;                       // [64,1024,1024] f32
  float* mask = dist + (size_t)BB * MM * NN;         // [64,1024,1024] as 0/1 f32

  float* xx   = (float*)d_ws;                        // 64*1024 floats
  float* yy   = xx + BB * MM;                        // 64*1024 floats
  float* sums = yy + BB * NN;                        // 64 floats

  // 1) row norms: (64*1024 + 64*1024) rows, 8 waves (8 rows) per block
  row_norms<<<(BB * MM + BB * NN) / 8, 256, 0, stream>>>(x, y, xx, yy);
  // 2) zero per-batch accumulators (ws is poisoned by harness)
  zero_sums<<<1, BB, 0, stream>>>(sums);
  // 3) WMMA distance GEMM: 128x64 tiles -> grid (16, 8, 64)
  dist_gemm<<<dim3(NN / TN, MM / TM, BB), 256, 0, stream>>>(x, y, xx, yy, dist, sums);
  // 4) mask pass: 64M elements, float4 per thread
  mask_kernel<<<(unsigned)((size_t)BB * MM * NN / 4 / 256), 256, 0, stream>>>(dist, sums, mask);
}